// Brain_connectomic_graph_12317966205115
// MI455X (gfx1250) — compile-verified
//
#include <hip/hip_runtime.h>
#include <hip/hip_bf16.h>
#include <math.h>

// ---- problem constants (match reference) ----
#define NN   100     // nodes
#define NE   2000    // edges
#define K1   50      // SAGPool keep
#define F1   64      // layer-1 width
#define F2   20      // layer-2 / cheb width
#define SLOPE 0.01f

typedef __attribute__((ext_vector_type(2))) float v2f;
typedef __attribute__((ext_vector_type(8))) float v8f;

// Generic fp32 WMMA matmul: C[M,N] (+)= A[M,K] @ B[K,N], all row-major.
// Uses V_WMMA_F32_16X16X4_F32. Tile loop is wave-uniform so EXEC is all-ones
// around every WMMA. All loads are UNCONDITIONAL (index clamped to a legal
// address, value zeroed by select) so no exec save/restore appears in the
// k-loop and loads can be clause-batched / overlapped with WMMA issue.
// Requires K % 4 == 0 (true at every call site: 100, 64, 20).
__device__ __forceinline__ void mm_wmma(const float* __restrict__ A, int lda,
                                        const float* __restrict__ B, int ldb,
                                        float* __restrict__ C, int ldc,
                                        int M, int N, int K, bool accum)
{
    const int lane = threadIdx.x & 31;
    const int wave = threadIdx.x >> 5;
    const int nw   = blockDim.x >> 5;
    const int half = lane >> 4;    // 0: lanes 0-15, 1: lanes 16-31
    const int l16  = lane & 15;
    const int mt = (M + 15) >> 4, nt = (N + 15) >> 4;

    for (int t = wave; t < mt * nt; t += nw) {      // wave-uniform
        const int m0 = (t / nt) << 4;
        const int n0 = (t % nt) << 4;

        const int  ar  = m0 + l16;                  // A row this lane owns
        const bool aok = ar < M;
        const int  arc = aok ? ar : (M - 1);        // clamped (always legal)
        const int  bcn = n0 + l16;                  // B col this lane owns
        const bool bok = bcn < N;
        const int  bcc = bok ? bcn : (N - 1);

        v8f acc = {0.f, 0.f, 0.f, 0.f, 0.f, 0.f, 0.f, 0.f};
        if (accum) {
            #pragma unroll
            for (int v = 0; v < 8; ++v) {
                int  r  = m0 + v + half * 8;
                bool rok = (r < M) && bok;
                int  rc = (r < M) ? r : (M - 1);
                float cv = C[rc * ldc + bcc];       // unconditional, legal
                acc[v] = rok ? cv : 0.f;
            }
        }

        const float* __restrict__ Arow = A + arc * lda + half * 2;
        const float* __restrict__ Bcol = B + (half * 2) * ldb + bcc;
        #pragma unroll 4
        for (int k = 0; k < K; k += 4) {
            float a0 = Arow[k];                     // contiguous pair, 8B aligned
            float a1 = Arow[k + 1];
            float b0 = Bcol[k * ldb];
            float b1 = Bcol[k * ldb + ldb];
            v2f a, b;
            a.x = aok ? a0 : 0.f;
            a.y = aok ? a1 : 0.f;
            b.x = bok ? b0 : 0.f;
            b.y = bok ? b1 : 0.f;
            acc = __builtin_amdgcn_wmma_f32_16x16x4_f32(
                      false, a, false, b, (short)0, acc, false, false);
        }
        #pragma unroll
        for (int v = 0; v < 8; ++v) {
            int r = m0 + v + half * 8, c = n0 + l16;
            if (r < M && c < N) C[r * ldc + c] = acc[v];
        }
    }
}

// self-loop + bias init for GCN rows [rlo,rhi): out = dis^2 * xw + b
__device__ __forceinline__ void gcn_init(const float* dis, const float* xw,
                                         const float* bias, float* out,
                                         int F, int rlo, int rhi)
{
    int cnt = (rhi - rlo) * F;
    for (int i = threadIdx.x; i < cnt; i += blockDim.x) {
        int n = rlo + i / F, f = i % F;
        out[n * F + f] = dis[n] * dis[n] * xw[n * F + f] + bias[f];
    }
}

// edge scatter: out[dst] += dis[src]*ew*dis[dst] * xw[src] for edges inside [lo,hi)
__device__ __forceinline__ void gcn_scatter(const int* ei, const float* ea,
                                            const float* dis, const float* xw,
                                            float* out, int F, int lo, int hi)
{
    for (int e = threadIdx.x; e < NE; e += blockDim.x) {
        int s = ei[e], d = ei[NE + e];
        if (s >= lo && s < hi && d >= lo && d < hi) {
            float w = dis[s] * ea[e] * dis[d];
            for (int f = 0; f < F; ++f)
                atomicAdd(&out[d * F + f], w * xw[s * F + f]);
        }
    }
}

__device__ __forceinline__ void lrelu_rows(float* buf, int F, int rlo, int rhi)
{
    int cnt = (rhi - rlo) * F;
    for (int i = threadIdx.x; i < cnt; i += blockDim.x) {
        float v = buf[rlo * F + i];
        buf[rlo * F + i] = v > 0.f ? v : SLOPE * v;
    }
}

__device__ __forceinline__ void rowsoftmax(const float* in, float* out, int rows)
{
    for (int n = threadIdx.x; n < rows; n += blockDim.x) {
        float mx = -1e30f;
        for (int f = 0; f < F2; ++f) mx = fmaxf(mx, in[n * F2 + f]);
        float sm = 0.f;
        float ev[F2];
        for (int f = 0; f < F2; ++f) { ev[f] = __expf(in[n * F2 + f] - mx); sm += ev[f]; }
        float inv = 1.f / sm;
        for (int f = 0; f < F2; ++f) out[n * F2 + f] = ev[f] * inv;
    }
}

__launch_bounds__(256)
__global__ void brain_graph_kernel(
    const float* __restrict__ x, const int* __restrict__ ei,
    const float* __restrict__ ea,
    const float* Wl1, const float* bl1, const float* Wr1, const float* br1,
    const float* Wl2, const float* bl2, const float* Wr2, const float* br2,
    const float* Wg1, const float* bg1, const float* Wrel, const float* brel,
    const float* Wroot, const float* Wc0, const float* Wc1, const float* Wc2,
    const float* bc, float* __restrict__ outp)
{
    // 51.2 KB reusable arena + small persistent arrays (~54 KB total)
    __shared__ float R[12800];
    __shared__ float dis_l[NN], dis_r[NN], dis_g[NN], dis_c[NN];
    __shared__ float score[NN], tsc[K1];
    __shared__ int   nidx[NN], perm[K1], sperm[K1];

    const int tid = threadIdx.x;

    // warm L2/WGP$ for the latency-critical first matmuls (global_prefetch_b8)
    {
        int n = tid;
        __builtin_prefetch(x + n * 40, 0, 0);           // 100x100 floats
        if (n < 100) __builtin_prefetch(Wl1 + n * 64, 0, 0);
        if (n < 100) __builtin_prefetch(Wr1 + n * 64, 0, 0);
    }

    float* XW   = R;          // [100,64] phase 1
    float* H1   = R + 6400;   // [100,64] phase 1-2
    float* XW2  = R;          // [100,20] phase 2
    float* H2C  = R + 2000;   // [100,20] phase 2 concat
    float* XWG  = R + 4000;   // [100,20] g1 stage
    float* H2   = R + 6400;   // [100,20] final node features (persists)
    float* AGG  = R;          // [100,20] SAG aggregate
    float* POOL = R + 2000;   // [50,20]  pooled (persists to end)
    float* TX1  = R + 3000;   // [100,20]
    float* TX2  = R + 8400;   // [100,20]
    float* SRAW = R + 10400;  // [100,20] s_raw -> ass (in place)
    float* S2   = R;          // [100,20]
    float* HC   = R + 5000;   // [20,20]
    float* INTER= R + 5400;   // [50,20]

    // ---- degrees (edge weights are fixed for both GCN layers) ----
    for (int n = tid; n < NN; n += blockDim.x) {
        dis_l[n] = 1.f; dis_r[n] = 1.f; dis_g[n] = 1.f;   // +1 self loop
    }
    __syncthreads();
    for (int e = tid; e < NE; e += blockDim.x) {
        int s = ei[e], d = ei[NE + e];
        float w = ea[e];
        if (s < 50 && d < 50)   atomicAdd(&dis_l[d], w);
        if (s >= 50 && d >= 50) atomicAdd(&dis_r[d], w);
        atomicAdd(&dis_g[d], w);
    }
    __syncthreads();
    for (int n = tid; n < NN; n += blockDim.x) {
        dis_l[n] = rsqrtf(dis_l[n]);
        dis_r[n] = rsqrtf(dis_r[n]);
        dis_g[n] = rsqrtf(dis_g[n]);
    }
    __syncthreads();

    // ---- layer 1 left: h_l rows [0,50) ----
    mm_wmma(x, NN, Wl1, F1, XW, F1, NN, F1, NN, false);
    __syncthreads();
    gcn_init(dis_l, XW, bl1, H1, F1, 0, 50);
    __syncthreads();
    gcn_scatter(ei, ea, dis_l, XW, H1, F1, 0, 50);
    __syncthreads();
    lrelu_rows(H1, F1, 0, 50);
    __syncthreads();
    // ---- layer 1 right: h_r rows [50,100) ----
    mm_wmma(x, NN, Wr1, F1, XW, F1, NN, F1, NN, false);
    __syncthreads();
    gcn_init(dis_r, XW, br1, H1, F1, 50, 100);
    __syncthreads();
    gcn_scatter(ei, ea, dis_r, XW, H1, F1, 50, 100);
    __syncthreads();
    lrelu_rows(H1, F1, 50, 100);
    __syncthreads();

    // ---- layer 2 left ----
    mm_wmma(H1, F1, Wl2, F2, XW2, F2, NN, F2, F1, false);
    __syncthreads();
    gcn_init(dis_l, XW2, bl2, H2C, F2, 0, 50);
    __syncthreads();
    gcn_scatter(ei, ea, dis_l, XW2, H2C, F2, 0, 50);
    __syncthreads();
    lrelu_rows(H2C, F2, 0, 50);
    __syncthreads();
    // ---- layer 2 right ----
    mm_wmma(H1, F1, Wr2, F2, XW2, F2, NN, F2, F1, false);
    __syncthreads();
    gcn_init(dis_r, XW2, br2, H2C, F2, 50, 100);
    __syncthreads();
    gcn_scatter(ei, ea, dis_r, XW2, H2C, F2, 50, 100);
    __syncthreads();
    lrelu_rows(H2C, F2, 50, 100);
    __syncthreads();

    // ---- g1 GCN over full graph (H1 dead; H2 overlays it) ----
    mm_wmma(H2C, F2, Wg1, F2, XWG, F2, NN, F2, F2, false);
    __syncthreads();
    gcn_init(dis_g, XWG, bg1, H2, F2, 0, 100);
    __syncthreads();
    gcn_scatter(ei, ea, dis_g, XWG, H2, F2, 0, 100);
    __syncthreads();
    lrelu_rows(H2, F2, 0, 100);
    __syncthreads();

    // ---- SAGPooling: unweighted GraphConv score ----
    for (int i = tid; i < NN * F2; i += blockDim.x) AGG[i] = 0.f;
    __syncthreads();
    for (int e = tid; e < NE; e += blockDim.x) {
        int s = ei[e], d = ei[NE + e];
        for (int f = 0; f < F2; ++f)
            atomicAdd(&AGG[d * F2 + f], H2[s * F2 + f]);
    }
    __syncthreads();
    for (int n = tid; n < NN; n += blockDim.x) {
        float acc = brel[0];
        for (int f = 0; f < F2; ++f)
            acc += AGG[n * F2 + f] * Wrel[f] + H2[n * F2 + f] * Wroot[f];
        score[n] = acc;
    }
    __syncthreads();
    // stable top-k rank (matches jax.lax.top_k tie-breaking)
    for (int n = tid; n < NN; n += blockDim.x) {
        int rk = 0;
        float sn = score[n];
        for (int m = 0; m < NN; ++m) {
            float sm = score[m];
            if (sm > sn || (sm == sn && m < n)) rk++;
        }
        if (rk < K1) { perm[rk] = n; tsc[rk] = sn; nidx[n] = rk; }
        else nidx[n] = -1;
    }
    __syncthreads();
    // sorted(perm) = selected node ids ascending
    for (int n = tid; n < NN; n += blockDim.x) {
        if (nidx[n] >= 0) {
            int p = 0;
            for (int m = 0; m < n; ++m) if (nidx[m] >= 0) p++;
            sperm[p] = n;
        }
    }
    __syncthreads();
    for (int i = tid; i < K1 * F2; i += blockDim.x) {
        int r = i / F2, f = i % F2;
        POOL[i] = H2[perm[r] * F2 + f] * tanhf(tsc[r]);
    }
    __syncthreads();

    // ---- ChebConv K=3 over relabeled pooled edges (unit weights) ----
    for (int n = tid; n < NN; n += blockDim.x) dis_c[n] = 0.f;   // deg first
    __syncthreads();
    for (int e = tid; e < NE; e += blockDim.x) {
        int pd = nidx[ei[NE + e]], ps = nidx[ei[e]];
        if (ps >= 0 && pd >= 0) atomicAdd(&dis_c[pd], 1.0f);
    }
    __syncthreads();
    for (int n = tid; n < NN; n += blockDim.x) {
        float dg = dis_c[n];
        dis_c[n] = dg > 0.f ? rsqrtf(fmaxf(dg, 1e-12f)) : 0.f;
    }
    __syncthreads();
    for (int i = tid; i < NN * F2; i += blockDim.x) { TX1[i] = 0.f; }
    __syncthreads();
    for (int e = tid; e < NE; e += blockDim.x) {
        int ps = nidx[ei[e]], pd = nidx[ei[NE + e]];
        if (ps >= 0 && pd >= 0) {
            float w = -dis_c[ps] * dis_c[pd];
            for (int f = 0; f < F2; ++f)
                atomicAdd(&TX1[pd * F2 + f], w * H2[ps * F2 + f]);
        }
    }
    __syncthreads();
    for (int i = tid; i < NN * F2; i += blockDim.x) { TX2[i] = 0.f; }
    __syncthreads();
    for (int e = tid; e < NE; e += blockDim.x) {
        int ps = nidx[ei[e]], pd = nidx[ei[NE + e]];
        if (ps >= 0 && pd >= 0) {
            float w = -dis_c[ps] * dis_c[pd];
            for (int f = 0; f < F2; ++f)
                atomicAdd(&TX2[pd * F2 + f], w * TX1[ps * F2 + f]);
        }
    }
    __syncthreads();
    for (int i = tid; i < NN * F2; i += blockDim.x)
        TX2[i] = 2.f * TX2[i] - H2[i];
    __syncthreads();
    // s_raw = Tx0@Wc0 + Tx1@Wc1 + Tx2@Wc2 + bc
    mm_wmma(H2,  F2, Wc0, F2, SRAW, F2, NN, F2, F2, false);
    __syncthreads();
    mm_wmma(TX1, F2, Wc1, F2, SRAW, F2, NN, F2, F2, true);
    __syncthreads();
    mm_wmma(TX2, F2, Wc2, F2, SRAW, F2, NN, F2, F2, true);
    __syncthreads();
    for (int i = tid; i < NN * F2; i += blockDim.x) SRAW[i] += bc[i % F2];
    __syncthreads();

    // ---- double softmax + diff-pool ----
    rowsoftmax(SRAW, SRAW, NN);      // ass (in place)
    __syncthreads();
    rowsoftmax(SRAW, S2, NN);        // s2
    __syncthreads();
    for (int t = tid; t < F2 * F2; t += blockDim.x) {   // H_coarse = s2.T @ h2
        int c = t / F2, f = t % F2;
        float acc = 0.f;
        for (int n = 0; n < NN; ++n) acc += S2[n * F2 + c] * H2[n * F2 + f];
        HC[t] = acc;
    }
    __syncthreads();
    for (int i = tid; i < K1 * F2; i += blockDim.x) {   // inter = ass[sort(perm)]
        int r = i / F2, f = i % F2;
        INTER[i] = SRAW[sperm[r] * F2 + f];
    }
    __syncthreads();
    // H2out = pooled + inter @ H_coarse  (accumulate into POOL)
    mm_wmma(INTER, F2, HC, F2, POOL, F2, K1, F2, F2, true);
    __syncthreads();
    for (int i = tid; i < K1 * F2; i += blockDim.x) outp[i] = POOL[i];
}

extern "C" void kernel_launch(void* const* d_in, const int* in_sizes, int n_in,
                              void* d_out, int out_size, void* d_ws, size_t ws_size,
                              hipStream_t stream)
{
    const float* x    = (const float*)d_in[0];
    const int*   ei   = (const int*)  d_in[1];
    const float* ea   = (const float*)d_in[2];
    // d_in[3] = adj (unused by reference)
    const float* Wl1  = (const float*)d_in[4];
    const float* bl1  = (const float*)d_in[5];
    const float* Wr1  = (const float*)d_in[6];
    const float* br1  = (const float*)d_in[7];
    const float* Wl2  = (const float*)d_in[8];
    const float* bl2  = (const float*)d_in[9];
    const float* Wr2  = (const float*)d_in[10];
    const float* br2  = (const float*)d_in[11];
    const float* Wg1  = (const float*)d_in[12];
    const float* bg1  = (const float*)d_in[13];
    const float* Wrel = (const float*)d_in[14];
    const float* brel = (const float*)d_in[15];
    const float* Wroot= (const float*)d_in[16];
    const float* Wc0  = (const float*)d_in[17];
    const float* Wc1  = (const float*)d_in[18];
    const float* Wc2  = (const float*)d_in[19];
    const float* bc   = (const float*)d_in[20];

    brain_graph_kernel<<<dim3(1), dim3(256), 0, stream>>>(
        x, ei, ea, Wl1, bl1, Wr1, br1, Wl2, bl2, Wr2, br2,
        Wg1, bg1, Wrel, brel, Wroot, Wc0, Wc1, Wc2, bc, (float*)d_out);
}